// UniformLaplacianSmoothnessLoss_1382979470101
// MI455X (gfx1250) — compile-verified
//
#include <hip/hip_runtime.h>

#define TILE 256

// Guaranteed hardware FP32 atomic add (non-returning global_atomic_add_f32),
// never a CAS loop. Accumulation order is unordered either way (matches the
// reference's segment_sum semantics); RNE rounding matches the L2 atomic unit.
__device__ __forceinline__ void fatomic_add(float* p, float v) {
    unsafeAtomicAdd(p, v);
}

// ---------------------------------------------------------------------------
// Pass 0: zero the accumulation workspace (deg + nbr_sum) every call.
// Vectorized b128 stores + scalar tail.
// ---------------------------------------------------------------------------
__global__ __launch_bounds__(256)
void ulsl_zero_kernel(float* __restrict__ ws, size_t n) {
    size_t n4 = n >> 2;
    float4* ws4 = (float4*)ws;
    size_t i = (size_t)blockIdx.x * blockDim.x + threadIdx.x;
    size_t stride = (size_t)gridDim.x * blockDim.x;
    float4 z = make_float4(0.f, 0.f, 0.f, 0.f);
    for (size_t j = i; j < n4; j += stride) ws4[j] = z;
    // tail
    for (size_t j = (n4 << 2) + i; j < n; j += stride) ws[j] = 0.0f;
}

// ---------------------------------------------------------------------------
// Pass 1: per-face scatter. Face indices are bulk-staged into LDS with the
// CDNA5 async global->LDS path (ASYNCcnt), then each thread handles one face
// for all 8 batches with the minimal atomic count:
//   deg[i]+=2, deg[j]+=2, deg[k]+=2               (3 atomics / face)
//   nbr[b,i] += (s - v_i) etc., s = v_i+v_j+v_k   (9 atomics / face / batch)
// ---------------------------------------------------------------------------
__global__ __launch_bounds__(256)
void ulsl_scatter_kernel(const float* __restrict__ vert,
                         const int*   __restrict__ faces,
                         float* __restrict__ deg,
                         float* __restrict__ nbr,
                         int N, int F) {
    __shared__ int s_faces[TILE * 3];

    const int t = threadIdx.x;
    const long long tileBase = (long long)blockIdx.x * (TILE * 3); // dword idx
    const long long totalDw  = (long long)F * 3;

    // Async-stage this tile's 768 face-index dwords into LDS (3 per thread).
    #pragma unroll
    for (int q = 0; q < 3; ++q) {
        long long d = tileBase + t + q * TILE;
        if (d > totalDw - 1) d = totalDw - 1;           // stay inside buffer
        const int* gptr = faces + d;
        unsigned ldsOff = (unsigned)(uintptr_t)(&s_faces[t + q * TILE]);
        asm volatile("global_load_async_to_lds_b32 %0, %1, off"
                     :: "v"(ldsOff), "v"(gptr)
                     : "memory");
    }
    asm volatile("s_wait_asynccnt 0" ::: "memory");
    __syncthreads();

    const int f = blockIdx.x * TILE + t;
    if (f >= F) return;

    const int vi = s_faces[t * 3 + 0];
    const int vj = s_faces[t * 3 + 1];
    const int vk = s_faces[t * 3 + 2];

    // Each face contributes 2 edges incident to each of its 3 vertices.
    fatomic_add(deg + vi, 2.0f);
    fatomic_add(deg + vj, 2.0f);
    fatomic_add(deg + vk, 2.0f);

    const size_t bi = (size_t)vi * 3;
    const size_t bj = (size_t)vj * 3;
    const size_t bk = (size_t)vk * 3;

    #pragma unroll
    for (int b = 0; b < 8; ++b) {
        const float* vb = vert + (size_t)b * N * 3;
        float ax = vb[bi + 0], ay = vb[bi + 1], az = vb[bi + 2];
        float bx = vb[bj + 0], by = vb[bj + 1], bz = vb[bj + 2];
        float cx = vb[bk + 0], cy = vb[bk + 1], cz = vb[bk + 2];
        float sx = ax + bx + cx;
        float sy = ay + by + cy;
        float sz = az + bz + cz;
        float* nb = nbr + (size_t)b * N * 3;
        fatomic_add(nb + bi + 0, sx - ax);
        fatomic_add(nb + bi + 1, sy - ay);
        fatomic_add(nb + bi + 2, sz - az);
        fatomic_add(nb + bj + 0, sx - bx);
        fatomic_add(nb + bj + 1, sy - by);
        fatomic_add(nb + bj + 2, sz - bz);
        fatomic_add(nb + bk + 0, sx - cx);
        fatomic_add(nb + bk + 1, sy - cy);
        fatomic_add(nb + bk + 2, sz - cz);
    }
}

// ---------------------------------------------------------------------------
// Pass 2: finalize. id = b*N + v, so nbr/vert share the base 3*id (coalesced).
// ---------------------------------------------------------------------------
__global__ __launch_bounds__(256)
void ulsl_finalize_kernel(const float* __restrict__ vert,
                          const float* __restrict__ deg,
                          const float* __restrict__ nbr,
                          float* __restrict__ out,
                          int N, long long total) {
    long long id = (long long)blockIdx.x * blockDim.x + threadIdx.x;
    if (id >= total) return;
    int v = (int)(id % (long long)N);
    float d    = fmaxf(deg[v], 1.0f);
    size_t base = (size_t)id * 3;
    float lx = nbr[base + 0] / d - vert[base + 0];
    float ly = nbr[base + 1] / d - vert[base + 1];
    float lz = nbr[base + 2] / d - vert[base + 2];
    out[id] = sqrtf(lx * lx + ly * ly + lz * lz);
}

extern "C" void kernel_launch(void* const* d_in, const int* in_sizes, int n_in,
                              void* d_out, int out_size, void* d_ws, size_t ws_size,
                              hipStream_t stream) {
    const float* vert  = (const float*)d_in[0];   // (B, N, 3) f32
    const int*   faces = (const int*)d_in[1];     // (F, 3) int

    const int B = 8;
    const int N = in_sizes[0] / (B * 3);
    const int F = in_sizes[1] / 3;

    float* out = (float*)d_out;                   // (B, N) f32
    float* deg = (float*)d_ws;                    // N floats
    float* nbr = deg + N;                         // B*N*3 floats

    // Pass 0: zero accumulators (deterministic across replays).
    size_t zn = (size_t)N + (size_t)B * N * 3;
    ulsl_zero_kernel<<<2048, 256, 0, stream>>>((float*)d_ws, zn);

    // Pass 1: face scatter with async-LDS index staging.
    int scatterBlocks = (F + TILE - 1) / TILE;
    ulsl_scatter_kernel<<<scatterBlocks, TILE, 0, stream>>>(vert, faces, deg, nbr, N, F);

    // Pass 2: finalize curvature.
    long long total = (long long)B * N;
    int finBlocks = (int)((total + 255) / 256);
    ulsl_finalize_kernel<<<finBlocks, 256, 0, stream>>>(vert, deg, nbr, out, N, total);
}